// GraphSAGE_2190433321456
// MI455X (gfx1250) — compile-verified
//
#include <hip/hip_runtime.h>
#include <hip/hip_bf16.h>
#include <math.h>

// ---------------------------------------------------------------------------
// GraphSAGE 2-layer (mean aggregate, normalize=True) for gfx1250 / MI455X.
//  - scatter phase: wave-per-edge, float4 per lane, global_atomic_add_f32
//  - GEMM phase   : V_WMMA_F32_16X16X4_F32, fused deg_inv scale, bias,
//                   L2-normalize, ReLU (and log_softmax in layer 2)
// ---------------------------------------------------------------------------

#define N_NODES  100000
#define N_EDGES  1600000
#define IN_DIM   128
#define HID_DIM  128
#define OUT_DIM  40
#define OUT_PAD  48        // 3 WMMA col-tiles of 16
#define EPS      1e-12f

typedef __attribute__((ext_vector_type(2))) float v2f;
typedef __attribute__((ext_vector_type(8))) float v8f;

__device__ __forceinline__ v8f wmma_f32(v2f a, v2f b, v8f c) {
    // D = A(16x4 f32) * B(4x16 f32) + C(16x16 f32)
    return __builtin_amdgcn_wmma_f32_16x16x4_f32(
        /*neg_a=*/false, a, /*neg_b=*/false, b,
        /*c_mod=*/(short)0, c, /*reuse_a=*/false, /*reuse_b=*/false);
}

// ----------------------------- utility kernels -----------------------------

__global__ void zero_f32_kernel(float* __restrict__ p, int n) {
    int i = blockIdx.x * blockDim.x + threadIdx.x;
    if (i < n) p[i] = 0.0f;
}

__global__ void degree_kernel(const long long* __restrict__ dst,
                              float* __restrict__ deg, int nedges) {
    int e = blockIdx.x * blockDim.x + threadIdx.x;
    if (e < nedges) atomicAdd(deg + dst[e], 1.0f);
}

__global__ void deg_inv_kernel(float* __restrict__ deg, int n) {
    int i = blockIdx.x * blockDim.x + threadIdx.x;
    if (i < n) deg[i] = 1.0f / fmaxf(deg[i], 1.0f);
}

// One wave32 per edge: 32 lanes x float4 = full 128-f32 feature row.
// Gathers and atomics both land in the 192MB L2 (x/h are 51.2MB each).
__global__ void scatter_add_kernel(const float* __restrict__ feat,
                                   const long long* __restrict__ src,
                                   const long long* __restrict__ dst,
                                   float* __restrict__ agg, int nedges) {
    int t = blockIdx.x * blockDim.x + threadIdx.x;
    int e = t >> 5;
    if (e >= nedges) return;
    int lane = threadIdx.x & 31;
    long long s = src[e];
    long long d = dst[e];
    const float4* sp = reinterpret_cast<const float4*>(feat + (size_t)s * IN_DIM);
    float4 v = sp[lane];
    float* ap = agg + (size_t)d * IN_DIM + (size_t)lane * 4;
    atomicAdd(ap + 0, v.x);
    atomicAdd(ap + 1, v.y);
    atomicAdd(ap + 2, v.z);
    atomicAdd(ap + 3, v.w);
}

// ----------------------------- layer 1 GEMM --------------------------------
// Block: 256 threads (8 waves). Block handles 16 node rows x 128 channels;
// each wave owns one 16x16 output tile. Epilogue: L2-normalize + ReLU.
__global__ void __launch_bounds__(256)
sage1_kernel(const float* __restrict__ agg,     // [N,128] neighbor sums
             const float* __restrict__ xin,     // [N,128]
             const float* __restrict__ dinv,    // [N]
             const float* __restrict__ Wl,      // [128,128] row-major
             const float* __restrict__ bias,    // [128]
             const float* __restrict__ Wr,      // [128,128]
             float* __restrict__ hout)          // [N,128]
{
    __shared__ float tile[16][HID_DIM + 4];
    __shared__ float inv_norm[16];

    const int tid  = threadIdx.x;
    const int wave = tid >> 5;          // 0..7 -> col tile
    const int lane = tid & 31;
    const int half = lane >> 4;         // 0/1
    const int m    = lane & 15;

    const int row0 = blockIdx.x * 16;   // 100000 % 16 == 0
    const int n0   = wave * 16;

    const int arow = row0 + m;          // A row this lane feeds
    const int bcol = n0 + m;            // output channel this lane feeds

    const float  dscale = dinv[arow];
    const float* ap = agg + (size_t)arow * IN_DIM;
    const float* xp = xin + (size_t)arow * IN_DIM;
    const float* wl = Wl  + (size_t)bcol * IN_DIM;
    const float* wr = Wr  + (size_t)bcol * IN_DIM;

    const float bv = bias[bcol];        // D(row, col=m): bias folds into acc
    v8f acc = {bv, bv, bv, bv, bv, bv, bv, bv};

    #pragma unroll 8
    for (int k0 = 0; k0 < IN_DIM; k0 += 4) {
        const int ka = k0 + 2 * half;
        float2 av = *reinterpret_cast<const float2*>(ap + ka);
        float2 xv = *reinterpret_cast<const float2*>(xp + ka);
        float2 lv = *reinterpret_cast<const float2*>(wl + ka);
        float2 rv = *reinterpret_cast<const float2*>(wr + ka);
        v2f a1 = {av.x * dscale, av.y * dscale};   // mean = sum * deg_inv
        v2f b1 = {lv.x, lv.y};
        acc = wmma_f32(a1, b1, acc);
        v2f a2 = {xv.x, xv.y};
        v2f b2 = {rv.x, rv.y};
        acc = wmma_f32(a2, b2, acc);
    }

    #pragma unroll
    for (int v = 0; v < 8; ++v)
        tile[v + 8 * half][n0 + m] = acc[v];
    __syncthreads();

    if (tid < 16) {
        float s = 0.0f;
        #pragma unroll 4
        for (int c = 0; c < HID_DIM; ++c) {
            float t = tile[tid][c];
            s += t * t;
        }
        inv_norm[tid] = 1.0f / fmaxf(sqrtf(s), EPS);
    }
    __syncthreads();

    #pragma unroll
    for (int i = tid; i < 16 * HID_DIM; i += 256) {
        int r = i >> 7;
        int c = i & (HID_DIM - 1);
        float v = tile[r][c] * inv_norm[r];
        hout[(size_t)(row0 + r) * HID_DIM + c] = fmaxf(v, 0.0f);
    }
}

// ----------------------------- layer 2 GEMM --------------------------------
// Block: 128 threads (4 waves); waves 0..2 own the 3 col tiles (N padded
// 40->48, invalid channels masked by zero B/bias). Epilogue: L2-normalize,
// ReLU, log_softmax over 40 classes.
__global__ void __launch_bounds__(128)
sage2_kernel(const float* __restrict__ agg,     // [N,128] neighbor sums of h
             const float* __restrict__ hin,     // [N,128]
             const float* __restrict__ dinv,    // [N]
             const float* __restrict__ Wl,      // [40,128]
             const float* __restrict__ bias,    // [40]
             const float* __restrict__ Wr,      // [40,128]
             float* __restrict__ out)           // [N,40] log-probs
{
    __shared__ float tile[16][OUT_PAD];

    const int tid  = threadIdx.x;
    const int wave = tid >> 5;
    const int lane = tid & 31;
    const int half = lane >> 4;
    const int m    = lane & 15;
    const int row0 = blockIdx.x * 16;

    if (wave < 3) {                       // wave-uniform: EXEC all-ones inside
        const int n0   = wave * 16;
        const int bcol = n0 + m;
        const float vm = (bcol < OUT_DIM) ? 1.0f : 0.0f;
        const int wrow = (bcol < OUT_DIM) ? bcol : 0;   // safe address

        const int arow = row0 + m;
        const float  dscale = dinv[arow];
        const float* ap = agg + (size_t)arow * HID_DIM;
        const float* xp = hin + (size_t)arow * HID_DIM;
        const float* wl = Wl  + (size_t)wrow * HID_DIM;
        const float* wr = Wr  + (size_t)wrow * HID_DIM;

        const float bv = bias[wrow] * vm;
        v8f acc = {bv, bv, bv, bv, bv, bv, bv, bv};

        #pragma unroll 8
        for (int k0 = 0; k0 < HID_DIM; k0 += 4) {
            const int ka = k0 + 2 * half;
            float2 av = *reinterpret_cast<const float2*>(ap + ka);
            float2 xv = *reinterpret_cast<const float2*>(xp + ka);
            float2 lv = *reinterpret_cast<const float2*>(wl + ka);
            float2 rv = *reinterpret_cast<const float2*>(wr + ka);
            v2f a1 = {av.x * dscale, av.y * dscale};
            v2f b1 = {lv.x * vm, lv.y * vm};
            acc = wmma_f32(a1, b1, acc);
            v2f a2 = {xv.x, xv.y};
            v2f b2 = {rv.x * vm, rv.y * vm};
            acc = wmma_f32(a2, b2, acc);
        }

        #pragma unroll
        for (int v = 0; v < 8; ++v)
            tile[v + 8 * half][n0 + m] = acc[v];
    }
    __syncthreads();

    // Per-row tail: normalize -> relu -> log_softmax (16 rows, 40 cols each)
    if (tid < 16) {
        const int r = tid;
        float s = 0.0f;
        #pragma unroll 4
        for (int c = 0; c < OUT_DIM; ++c) {
            float t = tile[r][c];
            s += t * t;
        }
        const float inv_n = 1.0f / fmaxf(sqrtf(s), EPS);

        float mx = -INFINITY;
        #pragma unroll 4
        for (int c = 0; c < OUT_DIM; ++c) {
            float z = fmaxf(tile[r][c] * inv_n, 0.0f);   // normalize + relu
            tile[r][c] = z;
            mx = fmaxf(mx, z);
        }
        float se = 0.0f;
        #pragma unroll 4
        for (int c = 0; c < OUT_DIM; ++c)
            se += expf(tile[r][c] - mx);
        const float lse = mx + logf(se);

        float* op = out + (size_t)(row0 + r) * OUT_DIM;
        #pragma unroll 4
        for (int c = 0; c < OUT_DIM; ++c)
            op[c] = tile[r][c] - lse;
    }
}

// ------------------------------- launcher ----------------------------------

extern "C" void kernel_launch(void* const* d_in, const int* in_sizes, int n_in,
                              void* d_out, int out_size, void* d_ws, size_t ws_size,
                              hipStream_t stream) {
    (void)in_sizes; (void)n_in; (void)out_size; (void)ws_size;

    const float*     x    = (const float*)d_in[0];
    const long long* ei   = (const long long*)d_in[1];   // int64 [2, E]
    const float*     W1l  = (const float*)d_in[2];
    const float*     b1   = (const float*)d_in[3];
    const float*     W1r  = (const float*)d_in[4];
    const float*     W2l  = (const float*)d_in[5];
    const float*     b2   = (const float*)d_in[6];
    const float*     W2r  = (const float*)d_in[7];
    float*           out  = (float*)d_out;

    const long long* src = ei;
    const long long* dst = ei + N_EDGES;

    // Workspace layout (floats): deg_inv | agg | h   (~103 MB total)
    float* ws  = (float*)d_ws;
    float* deg = ws;                                 // 100000 (pad to 102400)
    float* agg = ws + 102400;                        // 12.8M
    float* h   = agg + (size_t)N_NODES * HID_DIM;    // 12.8M

    const int featN = N_NODES * HID_DIM;             // 12,800,000
    const int B = 256;

    // degree -> deg_inv
    zero_f32_kernel<<<(N_NODES + B - 1) / B, B, 0, stream>>>(deg, N_NODES);
    degree_kernel<<<(N_EDGES + B - 1) / B, B, 0, stream>>>(dst, deg, N_EDGES);
    deg_inv_kernel<<<(N_NODES + B - 1) / B, B, 0, stream>>>(deg, N_NODES);

    // layer 1: aggregate x, then WMMA GEMM + normalize + relu -> h
    zero_f32_kernel<<<(featN + B - 1) / B, B, 0, stream>>>(agg, featN);
    scatter_add_kernel<<<(N_EDGES * 32) / B, B, 0, stream>>>(x, src, dst, agg, N_EDGES);
    sage1_kernel<<<N_NODES / 16, 256, 0, stream>>>(agg, x, deg, W1l, b1, W1r, h);

    // layer 2: aggregate h, then WMMA GEMM + normalize + relu + log_softmax
    zero_f32_kernel<<<(featN + B - 1) / B, B, 0, stream>>>(agg, featN);
    scatter_add_kernel<<<(N_EDGES * 32) / B, B, 0, stream>>>(h, src, dst, agg, N_EDGES);
    sage2_kernel<<<N_NODES / 16, 128, 0, stream>>>(agg, h, deg, W2l, b2, W2r, out);
}